// CRF_43344809951247
// MI455X (gfx1250) — compile-verified
//
#include <hip/hip_runtime.h>
#include <hip/hip_bf16.h>

#define START_IX 62
#define END_IX   63
#define NEGV     (-10000.0f)

#define SS 1024   // sequence length
#define TT 64     // number of tags

// ---- CDNA5 feature detection (device pass only) ------------------------
#if defined(__HIP_DEVICE_COMPILE__)
#if __has_builtin(__builtin_amdgcn_global_load_async_to_lds_b64)
#define HAVE_ASYNC_LDS 1
#endif
#endif

// ---- helpers -----------------------------------------------------------
// Pack a 6-bit index into the low mantissa bits of an f32 score so that a
// plain float max also carries the argmax. Low-6-bit perturbation is ~2^-18
// relative -- irrelevant for argmax over randn-scale scores.
__device__ __forceinline__ float pack_idx(float v, int idx) {
  return __int_as_float((__float_as_int(v) & 0xFFFFFFC0) | idx);
}
__device__ __forceinline__ float clear_idx(float v) {
  return __int_as_float(__float_as_int(v) & 0xFFFFFFC0);
}

__device__ __forceinline__ float lane_bcast(float v, int lane) {
#if defined(__HIP_DEVICE_COMPILE__)
  // v_readlane_b32: SGPR broadcast, no LDS traffic
  return __int_as_float(__builtin_amdgcn_readlane(__float_as_int(v), lane));
#else
  (void)lane; return v;
#endif
}
__device__ __forceinline__ int uniform_i32(int v) {
#if defined(__HIP_DEVICE_COMPILE__)
  return __builtin_amdgcn_readfirstlane(v);
#else
  return v;
#endif
}

#if HAVE_ASYNC_LDS
// Builtin signature (from hipcc diagnostics): params are v2i* in hardware
// address spaces (AS1 global src, AS3 LDS dst), then imm offset + cpol.
typedef int v2i32 __attribute__((vector_size(8)));
typedef __attribute__((address_space(1))) v2i32 gv2i32;
typedef __attribute__((address_space(3))) v2i32 lv2i32;

// Async global->LDS copy, 8 bytes per lane (ASYNCcnt path, gfx1250).
__device__ __forceinline__ void async_copy_b64(const void* g, const void* l) {
  __builtin_amdgcn_global_load_async_to_lds_b64(
      (gv2i32*)(unsigned long long)g,
      (lv2i32*)(unsigned)(unsigned long long)l,  // low 32 bits of generic = LDS addr
      0, 0);
}
__device__ __forceinline__ void wait_async_le1() {
#if __has_builtin(__builtin_amdgcn_s_wait_asynccnt)
  __builtin_amdgcn_s_wait_asynccnt(1);
#else
  asm volatile("s_wait_asynccnt 0x1" ::: "memory");
#endif
}
__device__ __forceinline__ void wait_async_le0() {
#if __has_builtin(__builtin_amdgcn_s_wait_asynccnt)
  __builtin_amdgcn_s_wait_asynccnt(0);
#else
  asm volatile("s_wait_asynccnt 0x0" ::: "memory");
#endif
}
#endif

// ---- main kernel: one wave per batch element ---------------------------
__global__ __launch_bounds__(32) void viterbi_crf_kernel(
    const float* __restrict__ feats,   // [B,S,T]
    const float* __restrict__ trans,   // [T,T]  (next, prev)
    float* __restrict__ score_out,     // [B]
    float* __restrict__ path_out)      // [B,S]
{
  __shared__ unsigned char bp[SS * TT];         // 64 KB backpointers (LDS!)
  __shared__ float         featBuf[2 * TT];     // async double buffer
  __shared__ float         tagsBuf[SS];         // staged path for coalesced dump

  const int l = threadIdx.x;            // 0..31 (wave32)
  const int b = blockIdx.x;
  const float* featBase = feats + (size_t)b * SS * TT;

  // Transitions are loop-invariant across all 1024 steps: lane l only ever
  // needs rows `l` and `l+32` of trans (next = l / l+32, all prev).
  // Hold them in 128 VGPRs -- zero DS traffic in the hot loop.
  float trA[TT];   // trA[k] = trans[l    *64 + k]
  float trB[TT];   // trB[k] = trans[(l+32)*64 + k]
#pragma unroll
  for (int k = 0; k < TT; k += 4) {
    float4 a = *(const float4*)&trans[l * TT + k];          // 256B-aligned rows
    float4 c = *(const float4*)&trans[(l + 32) * TT + k];
    trA[k] = a.x; trA[k + 1] = a.y; trA[k + 2] = a.z; trA[k + 3] = a.w;
    trB[k] = c.x; trB[k + 1] = c.y; trB[k + 2] = c.z; trB[k + 3] = c.w;
  }

  // fv distributed: lane l holds fv[l] (A) and fv[l+32] (B)
  float fvA = (l == START_IX)        ? 0.0f : NEGV;
  float fvB = ((l + 32) == START_IX) ? 0.0f : NEGV;

#if HAVE_ASYNC_LDS
  async_copy_b64(featBase + 2 * l, &featBuf[2 * l]);   // prologue: s=0
#endif

  for (int s = 0; s < SS; ++s) {
#if HAVE_ASYNC_LDS
    if (s + 1 < SS) {
      async_copy_b64(featBase + (size_t)(s + 1) * TT + 2 * l,
                     &featBuf[((s + 1) & 1) * TT + 2 * l]);
      wait_async_le1();                 // slot for step s has landed
    } else {
      wait_async_le0();
    }
    const int cur = (s & 1) * TT;
    float fA = *(volatile const float*)&featBuf[cur + l];
    float fB = *(volatile const float*)&featBuf[cur + l + 32];
#else
    if (s + 1 < SS) __builtin_prefetch(featBase + (size_t)(s + 1) * TT + 2 * l, 0, 0);
    float fA = featBase[(size_t)s * TT + l];
    float fB = featBase[(size_t)s * TT + l + 32];
#endif

    // max-plus reduction over prev, 2 independent max chains per output for ILP
    float a0 = -3.0e38f, a1 = -3.0e38f, b0 = -3.0e38f, b1 = -3.0e38f;
#pragma unroll
    for (int k = 0; k < TT; ++k) {
      float fk = lane_bcast((k < 32) ? fvA : fvB, k & 31);  // v_readlane_b32
      float vA = pack_idx(trA[k] + fk, k);                  // add + and_or
      float vB = pack_idx(trB[k] + fk, k);
      if (k & 1) { a1 = fmaxf(a1, vA); b1 = fmaxf(b1, vB); }
      else       { a0 = fmaxf(a0, vA); b0 = fmaxf(b0, vB); }
    }
    float bestA = fmaxf(a0, a1);
    float bestB = fmaxf(b0, b1);

    // record backpointers (argmax rides in the low 6 bits)
    bp[s * TT + l]      = (unsigned char)(__float_as_int(bestA) & 63);
    bp[s * TT + l + 32] = (unsigned char)(__float_as_int(bestB) & 63);

    fvA = clear_idx(bestA) + fA;
    fvB = clear_idx(bestB) + fB;
  }

  // terminal: fv + trans[END, prev], wave32 max+argmax reduce
  float kA = pack_idx(fvA + trans[END_IX * TT + l],       l);
  float kB = pack_idx(fvB + trans[END_IX * TT + l + 32],  l + 32);
  float m  = fmaxf(kA, kB);
#pragma unroll
  for (int off = 16; off >= 1; off >>= 1)
    m = fmaxf(m, __shfl_xor(m, off, 32));

  int btag = uniform_i32(__float_as_int(m) & 63);
  float fvt = (btag < 32) ? lane_bcast(fvA, btag & 31)
                          : lane_bcast(fvB, btag & 31);
  float score = fvt + trans[END_IX * TT + btag];
  if (l == 0) score_out[b] = score;

  // backtrack through LDS backpointers (serial, ~LDS latency per hop)
  int tag = btag;
  for (int s = SS - 1; s >= 0; --s) {
    tagsBuf[s] = (float)tag;
    tag = (int)bp[s * TT + tag];
  }

  // coalesced dump of the decoded path
  for (int i = l; i < SS; i += 32)
    path_out[(size_t)b * SS + i] = tagsBuf[i];
}

// ---- launch ------------------------------------------------------------
extern "C" void kernel_launch(void* const* d_in, const int* in_sizes, int n_in,
                              void* d_out, int out_size, void* d_ws, size_t ws_size,
                              hipStream_t stream) {
  (void)n_in; (void)out_size; (void)d_ws; (void)ws_size;
  const float* feats = (const float*)d_in[0];   // [B,S,T] f32
  const float* trans = (const float*)d_in[1];   // [T,T]   f32

  const int B = in_sizes[0] / (SS * TT);
  float* score = (float*)d_out;        // [B]
  float* path  = (float*)d_out + B;    // [B,S]

  viterbi_crf_kernel<<<dim3(B), dim3(32), 0, stream>>>(feats, trans, score, path);
}